// Model_89970974917394
// MI455X (gfx1250) — compile-verified
//
#include <hip/hip_runtime.h>
#include <hip/hip_bf16.h>
#include <math.h>

// ---------------------------------------------------------------------------
// Fused Informer-embedding + RBF-SVM head for MI455X (gfx1250, wave32).
//
//   enc_out = CircularConv1d(x_enc) + PosEmb + Linear(x_mark_enc)   [8,512,512]
//   K[n,s]  = exp(-g * ||x_n - sv_s||^2)                            [4096,50]
//   out     = K @ alphas + bias                                     [8,512,7]
//
// ||x-s||^2 = ||x||^2 + ||s||^2 - 2 x.s  ->  the x.s term is a
// [4096x512]x[512x64] GEMM done with V_WMMA_F32_16X16X4_F32 (f32 exact).
// Everything stays on-chip: X tile built in LDS, S chunks staged in LDS,
// no 8MB enc_out round trip to HBM.
// ---------------------------------------------------------------------------

typedef __attribute__((ext_vector_type(2))) float v2f;
typedef __attribute__((ext_vector_type(8))) float v8f;

#define BQ   8
#define LSEQ 512
#define CIN  7
#define DMOD 512
#define NSV  50
#define NSVP 64          // padded to 4 WMMA column tiles
#define GAMMA_F 0.1f

#define TILE_M   16      // rows of X per workgroup
#define KCHUNK   64      // K staged per iteration
#define XS_LD    514     // LDS stride for X tile (bank-spread, 8B aligned)
#define SS_LD    66      // LDS stride for S chunk

__global__ __launch_bounds__(128)
void svmformer_fused_kernel(const float* __restrict__ x_enc,       // [B,L,7]
                            const float* __restrict__ x_mark_enc,  // [B,L,4]
                            const float* __restrict__ W_token,     // [512,7,3]
                            const float* __restrict__ W_time,      // [512,4]
                            const float* __restrict__ SV,          // [50,512]
                            const float* __restrict__ alphas,      // [50]
                            const float* __restrict__ bias,        // [7]
                            float*       __restrict__ out)         // [B,L,7]
{
    __shared__ float Xs[TILE_M * XS_LD];    // 32896 B : embedded rows
    __shared__ float Ss[NSVP * SS_LD];      // 16896 B : support-vector K-chunk
    __shared__ float xe[18 * CIN];          // staged x_enc rows (with halo)
    __shared__ float xm[TILE_M * 4];        // staged x_mark_enc rows
    __shared__ float rn[TILE_M];            // ||x_row||^2
    __shared__ float sn[NSVP];              // ||sv||^2 (0 for padding)
    __shared__ float psum[TILE_M];          // sum_s K[n,s]*alpha[s]

    const int tid  = threadIdx.x;           // 0..127 (4 wave32s)
    const int wave = tid >> 5;              // 0..3 -> SV column tile
    const int lane = tid & 31;

    const int tile = blockIdx.x;            // 0..255
    const int n0   = tile * TILE_M;         // global row base (b*512 + l)
    const int b    = n0 >> 9;               // batch index (512 rows / batch)
    const int l0   = n0 & (LSEQ - 1);       // l range [l0, l0+16) within batch

    // ---- Phase 0: stage raw inputs for this tile -------------------------
    for (int i = tid; i < 18 * CIN; i += 128) {      // rows l0-1 .. l0+16 (wrap)
        int r = i / CIN, c = i % CIN;
        int l = (l0 - 1 + r) & (LSEQ - 1);
        xe[i] = x_enc[(b * LSEQ + l) * CIN + c];
    }
    for (int i = tid; i < TILE_M * 4; i += 128) {
        int r = i >> 2, f = i & 3;
        xm[i] = x_mark_enc[(b * LSEQ + l0 + r) * 4 + f];
    }
    __syncthreads();

    // ---- Phase 1: build embedded tile X[16,512] in LDS -------------------
    // X[row,d] = conv(circular,k=3) + pos_emb(l0+row,d) + time_linear
    for (int idx = tid; idx < TILE_M * DMOD; idx += 128) {
        int row = idx >> 9;                  // 0..15
        int d   = idx & (DMOD - 1);
        const float* wt = W_token + d * (CIN * 3);   // [c][k]
        float acc = 0.0f;
        #pragma unroll
        for (int c = 0; c < CIN; ++c) {
            #pragma unroll
            for (int k = 0; k < 3; ++k)      // src row (row+k) == l0+row-1+k
                acc += xe[(row + k) * CIN + c] * wt[c * 3 + k];
        }
        const float* wtm = W_time + d * 4;
        #pragma unroll
        for (int f = 0; f < 4; ++f) acc += xm[row * 4 + f] * wtm[f];
        // positional embedding: pe[l, 2i]=sin(l*div_i), pe[l,2i+1]=cos(l*div_i)
        int   i2  = d & ~1;                  // 2i
        float div = __expf((float)i2 * (-9.210340371976184f / 512.0f)); // -ln(1e4)/D
        float ang = (float)(l0 + row) * div;
        acc += (d & 1) ? __cosf(ang) : __sinf(ang);
        Xs[row * XS_LD + d] = acc;
    }
    __syncthreads();

    // ---- Phase 2: norms --------------------------------------------------
    if (tid < TILE_M) {
        float s = 0.0f;
        for (int d = 0; d < DMOD; ++d) { float v = Xs[tid * XS_LD + d]; s += v * v; }
        rn[tid] = s;
        psum[tid] = 0.0f;
    }
    if (tid < NSVP) {
        float s = 0.0f;
        if (tid < NSV)
            for (int d = 0; d < DMOD; ++d) { float v = SV[tid * DMOD + d]; s += v * v; }
        sn[tid] = s;
    }
    __syncthreads();

    // ---- Phase 3: G = X . S^T via V_WMMA_F32_16X16X4_F32 ----------------
    // A frag (16x4 f32): lane<16 -> {K=0,K=1}, lane>=16 -> {K=2,K=3}, M=lane&15
    // B frag (4x16 f32): same striping with N=lane&15 (ISA 7.12.2)
    const int mrow = lane & 15;
    const int kb   = (lane >> 4) << 1;       // 0 or 2
    v8f acc = {};                            // C/D: VGPR r -> M=r+8*(lane>=16)

    for (int kc = 0; kc < DMOD; kc += KCHUNK) {
        // stage padded S chunk [64 x 64] (zero rows for sv >= 50)
        for (int i = tid; i < NSVP * KCHUNK; i += 128) {
            int s = i >> 6, kk = i & (KCHUNK - 1);
            Ss[s * SS_LD + kk] = (s < NSV) ? SV[s * DMOD + kc + kk] : 0.0f;
        }
        __syncthreads();
        #pragma unroll
        for (int k4 = 0; k4 < KCHUNK; k4 += 4) {
            v2f a, bm;
            a.x  = Xs[mrow * XS_LD + kc + k4 + kb];
            a.y  = Xs[mrow * XS_LD + kc + k4 + kb + 1];
            bm.x = Ss[(wave * 16 + mrow) * SS_LD + k4 + kb];
            bm.y = Ss[(wave * 16 + mrow) * SS_LD + k4 + kb + 1];
            acc = __builtin_amdgcn_wmma_f32_16x16x4_f32(
                      false, a, false, bm, (short)0, acc, false, false);
        }
        __syncthreads();
    }

    // ---- Phase 4: RBF + alpha reduction ---------------------------------
    const int sv    = wave * 16 + (lane & 15);         // this lane's SV column
    const float svn = sn[sv];
    const float alp = (sv < NSV) ? alphas[sv] : 0.0f;  // padding contributes 0
    const int mhi   = (lane >> 4) * 8;
    #pragma unroll
    for (int r = 0; r < 8; ++r) {
        int   M     = r + mhi;
        float dist2 = rn[M] + svn - 2.0f * acc[r];
        float kv    = __expf(-GAMMA_F * dist2) * alp;
        atomicAdd(&psum[M], kv);                       // ds_add_f32
    }
    __syncthreads();

    // ---- Phase 5: broadcast + bias, write [16 x 7] output slab ----------
    for (int i = tid; i < TILE_M * CIN; i += 128) {
        int row = i / CIN, c = i % CIN;
        out[(n0 + row) * CIN + c] = psum[row] + bias[c];
    }
}

extern "C" void kernel_launch(void* const* d_in, const int* in_sizes, int n_in,
                              void* d_out, int out_size, void* d_ws, size_t ws_size,
                              hipStream_t stream) {
    (void)in_sizes; (void)n_in; (void)d_ws; (void)ws_size; (void)out_size;
    const float* x_enc      = (const float*)d_in[0];
    const float* x_mark_enc = (const float*)d_in[1];
    // d_in[2] = x_dec, d_in[3] = x_mark_dec : unused by the reference math
    const float* W_token    = (const float*)d_in[4];
    const float* W_time     = (const float*)d_in[5];
    const float* SV         = (const float*)d_in[6];
    const float* alphas     = (const float*)d_in[7];
    const float* bias       = (const float*)d_in[8];
    float*       out        = (float*)d_out;

    const int n_tiles = (BQ * LSEQ) / TILE_M;   // 4096/16 = 256 workgroups
    svmformer_fused_kernel<<<n_tiles, 128, 0, stream>>>(
        x_enc, x_mark_enc, W_token, W_time, SV, alphas, bias, out);
}